// DCNv2_1090921693819
// MI455X (gfx1250) — compile-verified
//
#include <hip/hip_runtime.h>

// DCNv2 fused deformable-im2col + f32 WMMA GEMM for gfx1250 (MI455X).
// x(B,256,64,64) f32, offset(B,18,64,64), mask(B,9,64,64),
// weight(256,256,3,3), bias(256) -> out(B,256,64,64) f32.
//
// Per workgroup (512 thr = 16 waves): one output row (64 px) x all 256 OC.
// K = 2304 reduced in 72 chunks of 32. Weights staged in LDS by the Tensor
// Data Mover (TDM) when available; sampled cols staged by VALU+gathers
// (x fits in the 192 MB L2, so bilinear corner gathers are L2-resident).

typedef __attribute__((ext_vector_type(2))) float v2f;
typedef __attribute__((ext_vector_type(8))) float v8f;
typedef __attribute__((ext_vector_type(4))) unsigned u32x4;
typedef __attribute__((ext_vector_type(4))) int i32x4;
typedef __attribute__((ext_vector_type(8))) int i32x8;

#define KH 3
#define KW 3
#define CIN 256
#define OCH 256
#define HH 64
#define WW 64
#define KTAP 9
#define KD (CIN * KTAP)        // 2304 = GEMM K
#define KC 32                  // K-chunk staged in LDS
#define NCHUNK (KD / KC)       // 72
#define NTILE 64               // pixels per workgroup = one output row
#define NWAVES 16
#define THREADS (NWAVES * 32)  // 512
#define WROW 36                // padded LDS row stride for weights (dwords)
#define CROW 34                // padded LDS row stride for cols (dwords, even)

#if defined(__has_builtin)
#if __has_builtin(__builtin_amdgcn_tensor_load_to_lds)
#define HAS_TDM 1
#endif
#endif

__device__ __forceinline__ unsigned lds_offset_of(const void* p) {
    // generic -> LDS addrspace(3) -> 32-bit LDS byte offset
    return (unsigned)(unsigned long long)(__attribute__((address_space(3))) const void*)p;
}

__global__ __launch_bounds__(THREADS)
void dcnv2_fused(const float* __restrict__ x,
                 const float* __restrict__ offset,
                 const float* __restrict__ mask,
                 const float* __restrict__ weight,
                 const float* __restrict__ bias,
                 float* __restrict__ out,
                 int B)
{
    // LDS: 36864 + 8704 + 9216 + 9216 = 64000 B (pool is 320 KB/WGP)
    __shared__ __align__(16) float wgt_lds[OCH * WROW];      // [oc][kd_local]
    __shared__ __align__(16) float cols_lds[NTILE * CROW];   // [n][kd_local]
    __shared__ __align__(16) int   tap_i[KTAP * NTILE * 4];  // 4 corner idx per (k,n)
    __shared__ __align__(16) float tap_w[KTAP * NTILE * 4];  // 4 corner wts (mask folded)

    const int oy   = blockIdx.x;
    const int b    = blockIdx.y;
    const int t    = threadIdx.x;
    const int lane = t & 31;
    const int l16  = lane & 15;
    const int half = lane >> 4;
    const int wave = t >> 5;
    const int m0   = wave * 16;

    // ---------------- precompute bilinear taps for this row ----------------
    for (int e = t; e < KTAP * NTILE; e += THREADS) {
        const int k  = e >> 6;
        const int n  = e & 63;
        const int ky = k / KW;
        const int kx = k - ky * KW;

        const float offy = offset[(((b * 2 * KTAP) + 2 * k + 0) * HH + oy) * WW + n];
        const float offx = offset[(((b * 2 * KTAP) + 2 * k + 1) * HH + oy) * WW + n];
        const float mval = mask[(((b * KTAP) + k) * HH + oy) * WW + n];

        const float py = (float)(oy - 1 + ky) + offy;   // STRIDE=1, PAD=1, DIL=1
        const float px = (float)(n - 1 + kx) + offx;

        const float y0f = floorf(py);
        const float x0f = floorf(px);
        const int   y0  = (int)y0f;
        const int   x0  = (int)x0f;
        const float wy1 = py - y0f, wx1 = px - x0f;
        const float wy[2] = {1.0f - wy1, wy1};
        const float wx[2] = {1.0f - wx1, wx1};
        const int   ys[2] = {y0, y0 + 1};
        const int   xs[2] = {x0, x0 + 1};

        int   idx[4];
        float w[4];
#pragma unroll
        for (int cy = 0; cy < 2; ++cy) {
#pragma unroll
            for (int cx = 0; cx < 2; ++cx) {
                const int  yy = ys[cy], xx = xs[cx];
                const bool valid = (yy >= 0) & (yy < HH) & (xx >= 0) & (xx < WW);
                const int  yi = min(max(yy, 0), HH - 1);
                const int  xi = min(max(xx, 0), WW - 1);
                idx[cy * 2 + cx] = yi * WW + xi;
                w[cy * 2 + cx]   = valid ? (wy[cy] * wx[cx] * mval) : 0.0f;
            }
        }
        ((int4*)tap_i)[e]   = make_int4(idx[0], idx[1], idx[2], idx[3]);
        ((float4*)tap_w)[e] = make_float4(w[0], w[1], w[2], w[3]);
    }
    __syncthreads();

    v8f acc[4] = {};

    const float* xb     = x + (size_t)b * CIN * HH * WW;
    const int    n_of_t = t & 63;
    const int    kdl0   = t >> 6;
#if !defined(HAS_TDM)
    const int    woc    = t >> 1;
    const int    wh     = t & 1;
#endif

    // deformable-sampled cols for one chunk -> registers
    auto sample_chunk = [&](int kc, float cv[4]) {
#pragma unroll
        for (int j = 0; j < 4; ++j) {
            const int kdl = kdl0 + 8 * j;
            const int kd  = kc + kdl;
            const int c   = kd / 9;
            const int k   = kd - c * 9;
            const int e   = (k << 6) | n_of_t;
            const int4   ii = ((const int4*)tap_i)[e];
            const float4 ww = ((const float4*)tap_w)[e];
            const float* p  = xb + c * (HH * WW);
            cv[j] = ww.x * p[ii.x] + ww.y * p[ii.y] + ww.z * p[ii.z] + ww.w * p[ii.w];
        }
    };

#if defined(HAS_TDM)
    // ---- TDM descriptor: 2D tile 32x256 dwords of the 2304x256 weight ----
    // Padding: 4 dwords after every 32 stored -> LDS row stride 36 (= WROW).
    const unsigned lds_base = lds_offset_of(wgt_lds);
    const unsigned g1d0 = (2u << 16)    // data_size = 4 bytes
                        | (1u << 20)    // pad_enable
                        | (4u << 22)    // pad_interval: 32 dwords
                        | (3u << 25);   // pad_amount:   4 dwords
    const i32x8 g1 = { (int)g1d0,
                       (int)((KD & 0xFFFFu) << 16),        // tensor_dim0 lo16 @bit48
                       (int)((unsigned)OCH << 16),         // dim0 hi16 | tensor_dim1 lo16
                       (int)((unsigned)KC << 16),          // dim1 hi16 | tile_dim0
                       (int)OCH,                           // tile_dim1 | tile_dim2=0
                       (int)KD,                            // tensor_dim0_stride[31:0]
                       0, 0 };
    const i32x4 gz = {0, 0, 0, 0};
#if __clang_major__ >= 23
    const i32x8 gz8 = {0, 0, 0, 0, 0, 0, 0, 0};
#endif
    auto tdm_load_weights = [&](int kc) {
        const unsigned long long ga = (unsigned long long)(const void*)(weight + kc);
        u32x4 g0;
        g0.x = 1u;                                   // count=1, no gather
        g0.y = lds_base;
        g0.z = (unsigned)(ga & 0xFFFFFFFFu);
        g0.w = (unsigned)((ga >> 32) & 0x01FFFFFFu) | (2u << 30);  // type=2
#if __clang_major__ >= 23
        __builtin_amdgcn_tensor_load_to_lds(g0, g1, gz, gz, gz8, 0);
#else
        __builtin_amdgcn_tensor_load_to_lds(g0, g1, gz, gz, 0);
#endif
    };
#endif

    // ---------------- software-pipelined main K loop -----------------------
    float cv[4];
    sample_chunk(0, cv);
#if !defined(HAS_TDM)
    float4 wv[4];
    {
        const float4* wp = (const float4*)(weight + woc * KD + wh * 16);
#pragma unroll
        for (int q = 0; q < 4; ++q) wv[q] = wp[q];
    }
#endif

#pragma unroll 1
    for (int ic = 0; ic < NCHUNK; ++ic) {
        const int kc = ic * KC;

        __syncthreads();   // WAR: previous chunk's WMMA reads complete

#if defined(HAS_TDM)
        if (wave == 0) tdm_load_weights(kc);        // DMA weights -> wgt_lds
#else
        {
            float4* ws = (float4*)&wgt_lds[woc * WROW + wh * 16];
#pragma unroll
            for (int q = 0; q < 4; ++q) ws[q] = wv[q];
        }
#endif
#pragma unroll
        for (int j = 0; j < 4; ++j)
            cols_lds[n_of_t * CROW + (kdl0 + 8 * j)] = cv[j];

        // prefetch next chunk while TDM/DS traffic is in flight
        if (ic + 1 < NCHUNK) {
            sample_chunk(kc + KC, cv);
#if !defined(HAS_TDM)
            const float4* wp = (const float4*)(weight + woc * KD + (kc + KC) + wh * 16);
#pragma unroll
            for (int q = 0; q < 4; ++q) wv[q] = wp[q];
#endif
        }

#if defined(HAS_TDM)
        if (wave == 0) __builtin_amdgcn_s_wait_tensorcnt(0);
#endif
        __syncthreads();   // RAW: staged chunk visible to all waves

        // 8 K-steps of v_wmma_f32_16x16x4_f32 across 4 N tiles
#pragma unroll
        for (int kk = 0; kk < KC; kk += 4) {
            const int ka = kk + 2 * half;   // VGPR0 holds K=0/2 by lane half
            v2f A = *((const v2f*)&wgt_lds[(m0 + l16) * WROW + ka]);
#pragma unroll
            for (int nt = 0; nt < 4; ++nt) {
                v2f Bf = *((const v2f*)&cols_lds[(nt * 16 + l16) * CROW + ka]);
                acc[nt] = __builtin_amdgcn_wmma_f32_16x16x4_f32(
                    false, A, false, Bf, (short)0, acc[nt], false, false);
            }
        }
    }

    // ---------------- epilogue ---------------------------------------------
#pragma unroll
    for (int nt = 0; nt < 4; ++nt) {
#pragma unroll
        for (int r = 0; r < 8; ++r) {
            const int oc = m0 + half * 8 + r;
            const int ox = nt * 16 + l16;
            out[(((size_t)b * OCH + oc) * HH + oy) * WW + ox] = acc[nt][r] + bias[oc];
        }
    }
}

extern "C" void kernel_launch(void* const* d_in, const int* in_sizes, int n_in,
                              void* d_out, int out_size, void* d_ws, size_t ws_size,
                              hipStream_t stream) {
    const float* x      = (const float*)d_in[0];
    const float* offset = (const float*)d_in[1];
    const float* mask   = (const float*)d_in[2];
    const float* weight = (const float*)d_in[3];
    const float* bias   = (const float*)d_in[4];
    float* out = (float*)d_out;

    const int B = in_sizes[0] / (CIN * HH * WW);
    dim3 grid(HH, B);
    dcnv2_fused<<<grid, THREADS, 0, stream>>>(x, offset, mask, weight, bias, out, B);
}